// QRMSNorm_30588757082116
// MI455X (gfx1250) — compile-verified
//
#include <hip/hip_runtime.h>
#include <stdint.h>

#define H_DIM 4096
#define BLOCK 256
#define ELEMS_PER_THREAD 16   // H_DIM / BLOCK
#define QMAX 127.0f
#define EPS 1e-6f

__device__ __forceinline__ float clip_round(float v) {
    return fminf(fmaxf(__builtin_rintf(v), -128.0f), 127.0f);
}

__global__ void __launch_bounds__(BLOCK)
qrmsnorm_kernel(const float* __restrict__ x,
                const float* __restrict__ w,
                float* __restrict__ q_out,
                float* __restrict__ scale_out,
                int rows) {
    __shared__ float sx[H_DIM];          // 16 KB staged row
    __shared__ float s_sum[8];
    __shared__ float s_max[8];
    __shared__ float s_bcast[2];         // [0] = inv quant factor, [1] = unused

    const int row = blockIdx.x;
    if (row >= rows) return;
    const int tid = threadIdx.x;

    const float* xrow = x + (size_t)row * H_DIM;

    // ---- Phase 1: async copy row (16 KB) global -> LDS via CDNA5 async path ----
    // Each of 256 threads issues 4 x b128 loads (GVS mode: saddr + 32-bit vgpr offset).
    {
        unsigned lds_base = (unsigned)(uintptr_t)(&sx[0]);
        unsigned byte_off = (unsigned)tid * 16u;
#pragma unroll
        for (int i = 0; i < 4; ++i) {
            unsigned go = byte_off + (unsigned)i * (BLOCK * 16u); // 4 KB stride per issue
            unsigned lo = lds_base + go;
            asm volatile("global_load_async_to_lds_b128 %0, %1, %2"
                         :: "v"(lo), "v"(go), "s"(xrow)
                         : "memory");
        }
        asm volatile("s_wait_asynccnt 0" ::: "memory");
    }
    __syncthreads();

    // ---- Phase 2: per-thread partial reductions over 16 contiguous elements ----
    const int base = tid * ELEMS_PER_THREAD;
    float4 xv[4], wv[4];
    float sumsq = 0.0f;
    float mx = 0.0f;
#pragma unroll
    for (int k = 0; k < 4; ++k) {
        xv[k] = *(const float4*)(&sx[base + 4 * k]);
        wv[k] = ((const float4*)w)[tid * 4 + k];
        float xw;
        sumsq += xv[k].x * xv[k].x;  xw = xv[k].x * wv[k].x;  mx = fmaxf(mx, fabsf(xw));
        sumsq += xv[k].y * xv[k].y;  xw = xv[k].y * wv[k].y;  mx = fmaxf(mx, fabsf(xw));
        sumsq += xv[k].z * xv[k].z;  xw = xv[k].z * wv[k].z;  mx = fmaxf(mx, fabsf(xw));
        sumsq += xv[k].w * xv[k].w;  xw = xv[k].w * wv[k].w;  mx = fmaxf(mx, fabsf(xw));
    }

    // ---- Phase 3: wave32 reduction, then cross-wave via LDS ----
#pragma unroll
    for (int off = 16; off > 0; off >>= 1) {
        sumsq += __shfl_xor(sumsq, off, 32);
        mx = fmaxf(mx, __shfl_xor(mx, off, 32));
    }
    const int wave = tid >> 5;           // 8 wave32s per block
    if ((tid & 31) == 0) {
        s_sum[wave] = sumsq;
        s_max[wave] = mx;
    }
    __syncthreads();

    if (tid == 0) {
        float tot = 0.0f, m = 0.0f;
#pragma unroll
        for (int i = 0; i < 8; ++i) {
            tot += s_sum[i];
            m = fmaxf(m, s_max[i]);
        }
        float var = tot * (1.0f / (float)H_DIM);
        float r = __frsqrt_rn(var + EPS);          // rsqrt
        float scale = m * r * (1.0f / QMAX);       // rms_scale = max|x*r*w| / 127
        scale_out[row] = scale;
        s_bcast[0] = QMAX / m;                     // r cancels: q = round(x*w*127/max|x*w|)
    }
    __syncthreads();

    const float inv = s_bcast[0];

    // ---- Phase 4: quantize and store (as float, per harness output convention) ----
    float4* qrow = (float4*)(q_out + (size_t)row * H_DIM);
#pragma unroll
    for (int k = 0; k < 4; ++k) {
        float4 o;
        o.x = clip_round(xv[k].x * wv[k].x * inv);
        o.y = clip_round(xv[k].y * wv[k].y * inv);
        o.z = clip_round(xv[k].z * wv[k].z * inv);
        o.w = clip_round(xv[k].w * wv[k].w * inv);
        qrow[tid * 4 + k] = o;
    }
}

extern "C" void kernel_launch(void* const* d_in, const int* in_sizes, int n_in,
                              void* d_out, int out_size, void* d_ws, size_t ws_size,
                              hipStream_t stream) {
    (void)n_in; (void)d_ws; (void)ws_size; (void)out_size;
    const float* x = (const float*)d_in[0];   // [B*S, H] fp32
    const float* w = (const float*)d_in[1];   // [H] fp32
    const int total = in_sizes[0];            // B*S*H
    const int rows = total / H_DIM;           // B*S = 16384

    float* q_out = (float*)d_out;                       // first B*S*H elements
    float* scale_out = q_out + (size_t)rows * H_DIM;    // then B*S scales

    qrmsnorm_kernel<<<rows, BLOCK, 0, stream>>>(x, w, q_out, scale_out, rows);
}